// LogSuperposition_2147483648611
// MI455X (gfx1250) — compile-verified
//
#include <hip/hip_runtime.h>
#include <math.h>

typedef __attribute__((ext_vector_type(2))) float v2f;
typedef __attribute__((ext_vector_type(8))) float v8f;

#define B_   4
#define F_   100
#define P_   16
#define WH   4096
#define FTILES 7

#define MS_OFFS 4.07866287f      /* sqrt(2*ln(4096)) */
#define LOG_NP  2.77258872f      /* ln(16)           */
#define EPS32   1.1920929e-07f

// ---------------------------------------------------------------------------
// Kernel 1: norms[b*16+p] = 0.5 * logsumexp_{xy}( 2*ms0(b,p,xy) )
// ---------------------------------------------------------------------------
__global__ void norm_kernel(const float* __restrict__ ms, float* __restrict__ norms) {
    int bp = blockIdx.x;                 // 0..63
    int b = bp >> 4, p = bp & 15;
    const float* src = ms + ((size_t)(b * 3 + 0) * P_ + p) * WH;
    int tid = threadIdx.x;               // 256 threads
    float mx = -INFINITY, s = 0.0f;
    for (int t = tid; t < WH; t += 256) {
        float v = 2.0f * src[t];
        if (v > mx) { s = s * expf(mx - v) + 1.0f; mx = v; }
        else        { s += expf(v - mx); }
    }
    __shared__ float sm[256], ss[256];
    sm[tid] = mx; ss[tid] = s;
    __syncthreads();
    for (int off = 128; off > 0; off >>= 1) {
        if (tid < off) {
            float m1 = sm[tid], s1 = ss[tid];
            float m2 = sm[tid + off], s2 = ss[tid + off];
            float m = fmaxf(m1, m2);
            sm[tid] = m;
            ss[tid] = s1 * expf(m1 - m) + s2 * expf(m2 - m);
        }
        __syncthreads();
    }
    if (tid == 0) norms[bp] = 0.5f * (sm[0] + logf(ss[0]));
}

// ---------------------------------------------------------------------------
// Kernel 2: per (b,f): Cmax = max_p tr0 ; Qr/Qi = exp(tr0-C) * t_hat ; bias(freq)
// ---------------------------------------------------------------------------
__global__ void qprep_kernel(const float* __restrict__ mr, const float* __restrict__ freqs,
                             float* __restrict__ Qr, float* __restrict__ Qi,
                             float* __restrict__ Cmax, float* __restrict__ bias) {
    int g = blockIdx.x * blockDim.x + threadIdx.x;
    if (g >= B_ * F_) return;
    const float* base = mr + (size_t)g * 3 * P_;
    float C = -INFINITY;
    for (int p = 0; p < P_; ++p) C = fmaxf(C, base[p]);
    for (int p = 0; p < P_; ++p) {
        float tr = base[P_ + p], ti = base[2 * P_ + p];
        float tn = sqrtf(fmaxf(tr * tr + ti * ti, EPS32));
        float e  = expf(base[p] - C);
        Qr[g * P_ + p] = e * tr / tn;
        Qi[g * P_ + p] = e * ti / tn;
    }
    Cmax[g] = C;
    // interp_bias: table[i] = cos(i*pi/127)*1.5 - 3
    float fr  = freqs[g];
    float pos = fminf(fmaxf((fr + 1.0f) * 0.5f * 127.0f, 0.0f), 127.0f);
    float lo  = floorf(pos);
    int ilo   = (int)lo;
    int ihi   = min(ilo + 1, 127);
    float w   = pos - lo;
    const float K = 3.14159265358979f / 127.0f;
    float tlo = cosf(ilo * K) * 1.5f - 3.0f;
    float thi = cosf(ihi * K) * 1.5f - 3.0f;
    bias[g] = tlo * (1.0f - w) + thi * w;
}

// ---------------------------------------------------------------------------
// Kernel 3: aux_sparse = mean_{b,p} softplus( 0.5*logsumexp_f( 2*tr0(b,f,p) ) )
// ---------------------------------------------------------------------------
__global__ void aux_kernel(const float* __restrict__ mr, float* __restrict__ out_aux) {
    int tid = threadIdx.x;               // 64 threads: one per (b,p)
    int b = tid >> 4, p = tid & 15;
    float mx = -INFINITY, s = 0.0f;
    for (int f = 0; f < F_; ++f) {
        float v = 2.0f * mr[((size_t)(b * F_ + f) * 3) * P_ + p];
        if (v > mx) { s = s * expf(mx - v) + 1.0f; mx = v; }
        else        { s += expf(v - mx); }
    }
    float c  = 0.5f * (mx + logf(s));
    float sp = fmaxf(c, 0.0f) + log1pf(expf(-fabsf(c)));   // stable softplus
    __shared__ float sh[64];
    sh[tid] = sp;
    __syncthreads();
    for (int off = 32; off > 0; off >>= 1) {
        if (tid < off) sh[tid] += sh[tid + off];
        __syncthreads();
    }
    if (tid == 0) out_aux[0] = sh[0] / 64.0f;
}

// ---------------------------------------------------------------------------
// Main kernel: one wave (32 threads) per 16-freq x 16-pixel tile.
// Complex GEMM  [SR|SI](f,x) = Q(f,p) * P(p,x)  via V_WMMA_F32_16X16X4_F32,
// then per-element log-sum-exp epilogue.
// ---------------------------------------------------------------------------
__global__ void __launch_bounds__(32) field_kernel(
    const float* __restrict__ ms,      // (B,3,P,WH)
    const float* __restrict__ mr,      // (B,F,3,P)
    const float* __restrict__ norms,   // (B,P)
    const float* __restrict__ Qr_g, const float* __restrict__ Qi_g,
    const float* __restrict__ Cmax_g, const float* __restrict__ bias_g,
    float* __restrict__ out)
{
    __shared__ float Pr[P_][16], Pi[P_][16], Ml[P_][16];   // [peak][pixel]
    __shared__ float Qr[16][P_], Qi[16][P_], T0[16][P_];   // [frow][peak]
    __shared__ float Cm[16], Bi[16];

    const int lane  = threadIdx.x;
    const int ptile = blockIdx.x;      // 0..255
    const int ftile = blockIdx.y;      // 0..6
    const int b     = blockIdx.z;      // 0..3
    const int x0    = ptile * 16;

    // ---- stage Q-side (A matrix: M=frow, K=peak) into LDS ----
    for (int t = 0; t < 8; ++t) {
        int idx = t * 32 + lane;
        int i = idx >> 4, p = idx & 15;
        int f = ftile * 16 + i;
        if (f < F_) {
            int g = b * F_ + f;
            Qr[i][p] = Qr_g[g * P_ + p];
            Qi[i][p] = Qi_g[g * P_ + p];
            T0[i][p] = mr[((size_t)g * 3) * P_ + p];
        } else {
            Qr[i][p] = 0.0f; Qi[i][p] = 0.0f; T0[i][p] = 0.0f;
        }
    }
    if (lane < 16) {
        int f = ftile * 16 + lane;
        int g = b * F_ + f;
        Cm[lane] = (f < F_) ? Cmax_g[g] : 0.0f;
        Bi[lane] = (f < F_) ? bias_g[g] : 0.0f;
    }

    // ---- stage P-side (B matrix: K=peak, N=pixel) into LDS ----
    for (int t = 0; t < 8; ++t) {
        int idx = t * 32 + lane;
        int p = idx >> 4, j = idx & 15;       // lanes 0..15 -> consecutive pixels
        size_t pix = (size_t)x0 + j;
        float a0  = ms[((size_t)(b * 3 + 0) * P_ + p) * WH + pix];
        float srv = ms[((size_t)(b * 3 + 1) * P_ + p) * WH + pix];
        float siv = ms[((size_t)(b * 3 + 2) * P_ + p) * WH + pix];
        float mln = a0 - norms[b * P_ + p] + MS_OFFS;
        float sn  = sqrtf(fmaxf(srv * srv + siv * siv, EPS32));
        float e   = expf(mln);
        Pr[p][j] = e * srv / sn;
        Pi[p][j] = e * siv / sn;
        Ml[p][j] = mln;
    }
    __syncthreads();

    // ---- complex GEMM: SR = Qr@Pr - Qi@Pi ; SI = Qr@Pi + Qi@Pr ----
    v8f aSR = {0.f,0.f,0.f,0.f,0.f,0.f,0.f,0.f};
    v8f aSI = {0.f,0.f,0.f,0.f,0.f,0.f,0.f,0.f};
    const int half = lane >> 4;     // 0: K={4k,4k+1}   1: K={4k+2,4k+3}
    const int l16  = lane & 15;
    for (int k = 0; k < 4; ++k) {
        int k0 = 4 * k + 2 * half;
        v2f aQr, aQi, aQiN, bPr, bPi;
        aQr.x = Qr[l16][k0];     aQr.y = Qr[l16][k0 + 1];
        aQi.x = Qi[l16][k0];     aQi.y = Qi[l16][k0 + 1];
        aQiN.x = -aQi.x;         aQiN.y = -aQi.y;          // f32 WMMA has no A/B neg
        bPr.x = Pr[k0][l16];     bPr.y = Pr[k0 + 1][l16];
        bPi.x = Pi[k0][l16];     bPi.y = Pi[k0 + 1][l16];
        aSR = __builtin_amdgcn_wmma_f32_16x16x4_f32(false, aQr,  false, bPr, (short)0, aSR, false, false);
        aSR = __builtin_amdgcn_wmma_f32_16x16x4_f32(false, aQiN, false, bPi, (short)0, aSR, false, false);
        aSI = __builtin_amdgcn_wmma_f32_16x16x4_f32(false, aQr,  false, bPi, (short)0, aSI, false, false);
        aSI = __builtin_amdgcn_wmma_f32_16x16x4_f32(false, aQi,  false, bPr, (short)0, aSI, false, false);
    }

    // ---- epilogue: per-element max, rescale, log-magnitude, direction ----
    float msv[16];
    #pragma unroll
    for (int p = 0; p < 16; ++p) msv[p] = Ml[p][l16];   // N (pixel) fixed per lane

    float* dir = out + (size_t)B_ * F_ * WH;            // field_dir base
    #pragma unroll
    for (int r = 0; r < 8; ++r) {
        int M = r + 8 * half;                           // C/D layout: lanes>=16 -> M+8
        int f = ftile * 16 + M;
        float m = -INFINITY;
        #pragma unroll
        for (int p = 0; p < 16; ++p) m = fmaxf(m, T0[M][p] + msv[p]);
        float scale = expf(Cm[M] - m);
        float sr = aSR[r] * scale;
        float si = aSI[r] * scale;
        float nrm = sqrtf(sr * sr + si * si + 0.001f);
        float lnm = m + logf(nrm);
        float outm = (lnm - LOG_NP) + Bi[M] * 0.5f;     // ((ln-log16)*2 + bias)/2
        float inv = 1.0f / nrm;
        if (f < F_) {
            size_t pix = (size_t)x0 + l16;
            size_t gf  = (size_t)(b * F_ + f);
            out[gf * WH + pix]             = outm;
            dir[(gf * 2 + 0) * WH + pix]   = sr * inv;
            dir[(gf * 2 + 1) * WH + pix]   = si * inv;
        }
    }
}

// ---------------------------------------------------------------------------
extern "C" void kernel_launch(void* const* d_in, const int* in_sizes, int n_in,
                              void* d_out, int out_size, void* d_ws, size_t ws_size,
                              hipStream_t stream) {
    const float* ms = (const float*)d_in[0];   // mode_shapes   (4,3,16,64,64)
    const float* mr = (const float*)d_in[1];   // mode_responses(4,100,3,16)
    const float* fr = (const float*)d_in[2];   // frequencies   (4,100)
    float* out = (float*)d_out;

    float* ws    = (float*)d_ws;
    float* norms = ws;                 // 64
    float* Qr    = norms + 64;         // 6400
    float* Qi    = Qr + B_ * F_ * P_;  // 6400
    float* Cmax  = Qi + B_ * F_ * P_;  // 400
    float* bias  = Cmax + B_ * F_;     // 400

    norm_kernel<<<B_ * P_, 256, 0, stream>>>(ms, norms);
    qprep_kernel<<<(B_ * F_ + 127) / 128, 128, 0, stream>>>(mr, fr, Qr, Qi, Cmax, bias);
    aux_kernel<<<1, 64, 0, stream>>>(mr, out + (size_t)B_ * F_ * WH * 3);

    dim3 grid(WH / 16, FTILES, B_);
    field_kernel<<<grid, 32, 0, stream>>>(ms, mr, norms, Qr, Qi, Cmax, bias, out);
}